// minSAD_Bilinaer_Affine_Registration_Layer_76020921139534
// MI455X (gfx1250) — compile-verified
//
#include <hip/hip_runtime.h>
#include <math.h>

#define IMG_W 1024
#define IMG_H 1024
#define NPAR  81
#define CHUNKS 128           // 128 chunks of 8 rows each
#define PIX_PER_CHUNK (IMG_W * IMG_H / CHUNKS)   // 8192
#define BLOCK 256
#define ITERS (PIX_PER_CHUNK / BLOCK)            // 32

typedef __attribute__((ext_vector_type(2))) float v2f;
typedef __attribute__((ext_vector_type(8))) float v8f;

__device__ __constant__ float c_SHIFTS[3] = {-1.0f, 0.0f, 1.0f};
__device__ __constant__ float c_ROTS[3]   = {-1.0f, 0.0f, 1.0f};
__device__ __constant__ float c_SCALES[3] = {1.0f - 1e-6f, 1.0f, 1.0f + 1e-6f};

// ds_swizzle group-of-32 xor patterns: offset = xor_mask<<10 | and_mask(0x1f)
__device__ __forceinline__ float swz_add(float v, float dummy) { return v; }

// Cross-lane reduction using the CDNA5 matrix pipe in full f32:
// A = ones(16x4), B holds lane partials in K=0 (lanes 0-15 -> cols, VGPR0)
// and K=2 (lanes 16-31 -> cols, VGPR0). D[m][n] = sum_k B[k][n] = s[n]+s[n+16].
// Then 4 single-instruction ds_swizzle xor-folds -> full wave32 sum.
__device__ __forceinline__ float wave_reduce_f32(float s) {
    v2f a; a[0] = 1.0f; a[1] = 1.0f;     // ones A-matrix
    v2f b; b[0] = s;    b[1] = 0.0f;     // partials in K=0 / K=2 slots
    v8f c = {};
    c = __builtin_amdgcn_wmma_f32_16x16x4_f32(
            /*neg_a=*/false, a, /*neg_b=*/false, b,
            /*c_mod=*/(short)0, c, /*reuse_a=*/false, /*reuse_b=*/false);
    float v = c[0];                      // lane n: s[n%16] + s[n%16 + 16]
    v += __int_as_float(__builtin_amdgcn_ds_swizzle(__float_as_int(v), 0x041F)); // SWAPX1
    v += __int_as_float(__builtin_amdgcn_ds_swizzle(__float_as_int(v), 0x081F)); // SWAPX2
    v += __int_as_float(__builtin_amdgcn_ds_swizzle(__float_as_int(v), 0x101F)); // SWAPX4
    v += __int_as_float(__builtin_amdgcn_ds_swizzle(__float_as_int(v), 0x201F)); // SWAPX8
    return v;                            // wave total (lane-uniform per 16-group)
}

__device__ __forceinline__ float tap(const float* __restrict__ img, int y, int x) {
    bool valid = (x >= 0) & (x < IMG_W) & (y >= 0) & (y < IMG_H);
    int xc = min(max(x, 0), IMG_W - 1);
    int yc = min(max(y, 0), IMG_H - 1);
    float v = img[yc * IMG_W + xc];
    return valid ? v : 0.0f;
}

__global__ void init_kernel(float* __restrict__ sums) {
    int t = blockIdx.x * blockDim.x + threadIdx.x;
    if (t < NPAR) sums[t] = 0.0f;
}

__global__ __launch_bounds__(BLOCK)
void sad_kernel(const float* __restrict__ img1, const float* __restrict__ img2,
                float* __restrict__ sums) {
    const int p     = blockIdx.x;   // parameter index 0..80
    const int chunk = blockIdx.y;   // 8-row slab
    const int t     = threadIdx.x;

    // Nested-loop order of the reference: sx outer ... scale inner.
    const int ixp = p / 27, iyp = (p / 9) % 3, irp = (p / 3) % 3, isp = p % 3;
    const float sx  = c_SHIFTS[ixp];
    const float sy  = c_SHIFTS[iyp];
    const float rot = c_ROTS[irp];
    const float sc  = c_SCALES[isp];

    const float ang  = rot * 0.017453292519943295f;   // deg -> rad
    const float cosv = cosf(ang) * sc;
    const float sinv = sinf(ang) * sc;
    const float tx   = sx * (2.0f / IMG_W);
    const float ty   = sy * (2.0f / IMG_H);

    float acc = 0.0f;
    const int base = chunk * PIX_PER_CHUNK + t;
    #pragma unroll 4
    for (int i = 0; i < ITERS; ++i) {
        const int g = base + i * BLOCK;
        const int h = g >> 10;
        const int w = g & (IMG_W - 1);

        // affine_grid base coords (align_corners=False)
        const float x = (2.0f * (float)w + 1.0f) * (1.0f / IMG_W) - 1.0f;
        const float y = (2.0f * (float)h + 1.0f) * (1.0f / IMG_H) - 1.0f;
        const float gx = cosv * x - sinv * y + tx;
        const float gy = sinv * x + cosv * y + ty;

        // grid_sample: bilinear, zero padding
        const float fx = (gx + 1.0f) * (IMG_W * 0.5f) - 0.5f;
        const float fy = (gy + 1.0f) * (IMG_H * 0.5f) - 0.5f;
        const float x0f = floorf(fx), y0f = floorf(fy);
        const float wx = fx - x0f,    wy = fy - y0f;
        const int   x0 = (int)x0f,    y0 = (int)y0f;

        const float v00 = tap(img1, y0,     x0);
        const float v01 = tap(img1, y0,     x0 + 1);
        const float v10 = tap(img1, y0 + 1, x0);
        const float v11 = tap(img1, y0 + 1, x0 + 1);

        const float warped = v00 * (1.0f - wx) * (1.0f - wy)
                           + v01 * wx          * (1.0f - wy)
                           + v10 * (1.0f - wx) * wy
                           + v11 * wx          * wy;

        acc += fabsf(warped - img2[g]);
    }

    // Wave32 reduction on the matrix pipe (EXEC is all-ones here: no divergence).
    const float wsum = wave_reduce_f32(acc);

    __shared__ float lds[BLOCK / 32];
    const int wave = t >> 5;
    const int lane = t & 31;
    if (lane == 0) lds[wave] = wsum;
    __syncthreads();
    if (t == 0) {
        float bsum = 0.0f;
        #pragma unroll
        for (int k = 0; k < BLOCK / 32; ++k) bsum += lds[k];
        atomicAdd(&sums[p], bsum);
    }
}

__global__ void finalize_kernel(const float* __restrict__ sums, float* __restrict__ out) {
    if (threadIdx.x != 0 || blockIdx.x != 0) return;

    const float inv = 1.0f / (float)(IMG_W * IMG_H);
    float S[NPAR];
    int best = 0;
    float bv;
    for (int i = 0; i < NPAR; ++i) S[i] = sums[i] * inv;
    bv = S[0];
    for (int i = 1; i < NPAR; ++i) {
        if (S[i] < bv) { bv = S[i]; best = i; }
    }
    const int ix = best / 27, iy = (best / 9) % 3, ir = (best / 3) % 3, is = best % 3;

    auto IDX = [](int a, int b, int c, int d) { return ((a * 3 + b) * 3 + c) * 3 + d; };
    auto FIT = [](float f0, float f1, float f2) {
        const float a = 0.5f * (f0 + f2) - f1;
        const float b = 0.5f * (f2 - f0);
        return -b / (2.0f * a);
    };

    // circular-wrap neighbors along each axis
    const float dx = FIT(S[IDX((ix + 2) % 3, iy, ir, is)], S[IDX(ix, iy, ir, is)],
                         S[IDX((ix + 1) % 3, iy, ir, is)]);
    const float dy = FIT(S[IDX(ix, (iy + 2) % 3, ir, is)], S[IDX(ix, iy, ir, is)],
                         S[IDX(ix, (iy + 1) % 3, ir, is)]);
    const float dr = FIT(S[IDX(ix, iy, (ir + 2) % 3, is)], S[IDX(ix, iy, ir, is)],
                         S[IDX(ix, iy, (ir + 1) % 3, is)]);
    const float ds = FIT(S[IDX(ix, iy, ir, (is + 2) % 3)], S[IDX(ix, iy, ir, is)],
                         S[IDX(ix, iy, ir, (is + 1) % 3)]);

    out[0] = c_SHIFTS[ix] + dx * (c_SHIFTS[2] - c_SHIFTS[1]);
    out[1] = c_SHIFTS[iy] + dy * (c_SHIFTS[2] - c_SHIFTS[1]);
    out[2] = c_ROTS[ir]   + dr * (c_ROTS[2]   - c_ROTS[1]);
    out[3] = c_SCALES[is] + ds * (c_SCALES[2] - c_SCALES[1]);
}

extern "C" void kernel_launch(void* const* d_in, const int* in_sizes, int n_in,
                              void* d_out, int out_size, void* d_ws, size_t ws_size,
                              hipStream_t stream) {
    const float* img1 = (const float*)d_in[0];
    const float* img2 = (const float*)d_in[1];
    float* out  = (float*)d_out;
    float* sums = (float*)d_ws;   // 81 f32 accumulators

    init_kernel<<<1, 128, 0, stream>>>(sums);
    sad_kernel<<<dim3(NPAR, CHUNKS), BLOCK, 0, stream>>>(img1, img2, sums);
    finalize_kernel<<<1, 32, 0, stream>>>(sums, out);
}